// GraphAttentionLayer_2293512536804
// MI455X (gfx1250) — compile-verified
//
#include <hip/hip_runtime.h>
#include <math.h>

typedef __attribute__((ext_vector_type(2))) float v2f;
typedef __attribute__((ext_vector_type(4))) float v4f;
typedef __attribute__((ext_vector_type(8))) float v8f;

#define N_NODES  50000
#define IN_DIM   256
#define OUT_DIM  256
#define HEADS    4
#define HEAD_DIM 64
#define NEG_SLOPE 0.2f

#define KCHUNK        64                 // k-values staged per LDS chunk (64KB)
#define ROWTILES      (N_NODES / 16)     // 3125 row tiles of 16 rows
#define WAVES_PER_WG  8

// ---------------------------------------------------------------------------
// Kernel 1: h = x @ W via V_WMMA_F32_16X16X4_F32, W staged through LDS.
//
// Workgroup = 8 waves; each wave computes a 16-row x 256-col tile of h.
// W is staged chunk-wise (64 k-rows = 64KB) in a quad-k interleaved layout:
//   Wlds[(k>>2)*1024 + col*4 + (k&3)]
// so each lane's two consecutive k-values are one aligned ds_load_b64, and
// half-wave bank usage is disjoint (banks {0,1} vs {2,3} mod 4) -> no
// conflicts.
//
// WMMA fragment layouts (wave32):
//  A (16x4 f32): lane L holds x[row0 + L%16][kb + 2*(L/16) + {0,1}]
//  B (4x16 f32): lane L holds W[kb + 2*(L/16) + {0,1}][col0 + L%16]
//  C/D:          VGPR j, lane L -> h[row0 + 8*(L/16) + j][col0 + L%16]
// ---------------------------------------------------------------------------
__global__ __launch_bounds__(256) void gat_gemm_wmma(
    const float* __restrict__ x, const float* __restrict__ W,
    float* __restrict__ h) {
  __shared__ float Wlds[16 * 1024];      // 64 k x 256 col, quad-k interleave

  const int tid  = threadIdx.x;
  const int wave = tid >> 5;
  const int lane = tid & 31;
  const int m    = lane & 15;
  const int g    = lane >> 4;            // half-wave: k-pair select
  const int tile = blockIdx.x * WAVES_PER_WG + wave;
  const bool active = (tile < ROWTILES); // wave-uniform guard (EXEC all-1s)
  const int row0 = tile * 16;

  v8f acc[16] = {};                      // 16 col-tiles x 8 VGPRs

  const float* xrow =
      active ? (x + (size_t)(row0 + m) * IN_DIM) : x;

  const int kl0 = tid >> 6;              // k_local & 3  (0..3)
  const int c0  = (tid & 63) * 4;        // column base for staging

  for (int kc = 0; kc < IN_DIM; kc += KCHUNK) {
    // ---- cooperative stage: W[kc..kc+63][0..255] -> LDS (interleaved) ----
    for (int i = 0; i < 16; ++i) {
      const int kglob = kc + i * 4 + kl0;
      const v4f wv = *(const v4f*)(W + (size_t)kglob * OUT_DIM + c0);
      float* dst = &Wlds[i * 1024 + c0 * 4 + kl0];
      dst[0]  = wv.x;
      dst[4]  = wv.y;
      dst[8]  = wv.z;
      dst[12] = wv.w;
    }
    __syncthreads();

    if (active) {
      for (int kb = 0; kb < KCHUNK; kb += 4) {
        // A fragment: aligned float2 from x
        const v2f a = *(const v2f*)(xrow + kc + kb + 2 * g);
        const float* Wp = &Wlds[(kb >> 2) * 1024 + m * 4 + 2 * g];
#pragma unroll
        for (int ct = 0; ct < 16; ++ct) {
          const v2f b = *(const v2f*)(Wp + ct * 64);  // ds_load_b64
          acc[ct] = __builtin_amdgcn_wmma_f32_16x16x4_f32(
              /*neg_a=*/false, a, /*neg_b=*/false, b,
              /*c_mod=*/(short)0, acc[ct], /*reuse_a=*/false,
              /*reuse_b=*/false);
        }
      }
    }
    __syncthreads();                     // before next chunk overwrites LDS
  }

  if (active) {
#pragma unroll
    for (int ct = 0; ct < 16; ++ct) {
#pragma unroll
      for (int j = 0; j < 8; ++j) {
        h[(size_t)(row0 + 8 * g + j) * OUT_DIM + ct * 16 + m] = acc[ct][j];
      }
    }
  }
}

// ---------------------------------------------------------------------------
// Kernel 2: a_src[n,h], a_dst[n,h] = <h[n,h,:], att_src[h]>, <h[n,h,:], att_dst[h]>
// Also initializes the segment-max (-inf) and denom (0) buffers.
// ---------------------------------------------------------------------------
__global__ void gat_attn_coef(const float* __restrict__ h,
                              const float* __restrict__ att_src,
                              const float* __restrict__ att_dst,
                              float* __restrict__ asrc, float* __restrict__ adst,
                              float* __restrict__ mmax, float* __restrict__ denom) {
  int t = blockIdx.x * blockDim.x + threadIdx.x;
  if (t >= N_NODES * HEADS) return;
  const int n  = t >> 2;
  const int hh = t & 3;
  const float* hv = h + (size_t)n * OUT_DIM + hh * HEAD_DIM;
  const float* as = att_src + hh * HEAD_DIM;
  const float* ad = att_dst + hh * HEAD_DIM;
  float s = 0.f, d = 0.f;
#pragma unroll 8
  for (int i = 0; i < HEAD_DIM; ++i) {
    s = fmaf(hv[i], as[i], s);
    d = fmaf(hv[i], ad[i], d);
  }
  asrc[t]  = s;
  adst[t]  = d;
  mmax[t]  = -__builtin_inff();
  denom[t] = 0.f;
}

// ---------------------------------------------------------------------------
// Kernel 3: out[n, :] = bias (fresh every call; harness poisons d_out)
// ---------------------------------------------------------------------------
__global__ void gat_out_init(const float* __restrict__ bias,
                             float* __restrict__ out) {
  int t = blockIdx.x * blockDim.x + threadIdx.x;
  if (t >= N_NODES * OUT_DIM) return;
  out[t] = bias[t & (OUT_DIM - 1)];
}

// float atomic max via signed-max / unsigned-min bit trick (init = -inf)
__device__ __forceinline__ void atomicMaxF(float* addr, float val) {
  if (val >= 0.f)
    atomicMax((int*)addr, __float_as_int(val));
  else
    atomicMin((unsigned int*)addr, __float_as_uint(val));
}

__device__ __forceinline__ void edge_endpoints(const int* __restrict__ ei,
                                               int E, int e, int& s, int& d) {
  if (e < E) { s = ei[e]; d = ei[E + e]; }
  else       { s = d = e - E; }          // self-loops appended
}

__device__ __forceinline__ float edge_logit(const float* __restrict__ asrc,
                                            const float* __restrict__ adst,
                                            int s, int d, int hh) {
  float v = asrc[s * HEADS + hh] + adst[d * HEADS + hh];
  return v > 0.f ? v : NEG_SLOPE * v;    // leaky_relu
}

// ---------------------------------------------------------------------------
// Kernel 4: segment max over edges per (dst, head)
// ---------------------------------------------------------------------------
__global__ void gat_edge_max(const int* __restrict__ ei, int E,
                             const float* __restrict__ asrc,
                             const float* __restrict__ adst,
                             float* __restrict__ mmax) {
  int t = blockIdx.x * blockDim.x + threadIdx.x;
  const int total = (E + N_NODES) * HEADS;
  if (t >= total) return;
  const int e = t >> 2, hh = t & 3;
  int s, d;
  edge_endpoints(ei, E, e, s, d);
  atomicMaxF(&mmax[d * HEADS + hh], edge_logit(asrc, adst, s, d, hh));
}

// ---------------------------------------------------------------------------
// Kernel 5: denom[dst,h] = sum exp(e - max)
// ---------------------------------------------------------------------------
__global__ void gat_edge_sum(const int* __restrict__ ei, int E,
                             const float* __restrict__ asrc,
                             const float* __restrict__ adst,
                             const float* __restrict__ mmax,
                             float* __restrict__ denom) {
  int t = blockIdx.x * blockDim.x + threadIdx.x;
  const int total = (E + N_NODES) * HEADS;
  if (t >= total) return;
  const int e = t >> 2, hh = t & 3;
  int s, d;
  edge_endpoints(ei, E, e, s, d);
  const float v = edge_logit(asrc, adst, s, d, hh);
  atomicAdd(&denom[d * HEADS + hh], expf(v - mmax[d * HEADS + hh]));
}

// ---------------------------------------------------------------------------
// Kernel 6: out[dst] += alpha * h[src].  64 threads per edge, float4 per
// thread: the 1KB h[src] gather and the 1KB out[dst] atomic burst are each
// fully contiguous & cacheline aligned.
// ---------------------------------------------------------------------------
__global__ void gat_edge_scatter(const int* __restrict__ ei, int E,
                                 const float* __restrict__ h,
                                 const float* __restrict__ asrc,
                                 const float* __restrict__ adst,
                                 const float* __restrict__ mmax,
                                 const float* __restrict__ denom,
                                 float* __restrict__ out) {
  int t = blockIdx.x * blockDim.x + threadIdx.x;
  const int total = (E + N_NODES) * 64;
  if (t >= total) return;
  const int e = t >> 6;
  const int l = t & 63;                 // covers 256 floats as float4
  const int hh = l >> 4;                // head = (l*4)/64
  int s, d;
  edge_endpoints(ei, E, e, s, d);
  const float v = edge_logit(asrc, adst, s, d, hh);
  const float alpha =
      expf(v - mmax[d * HEADS + hh]) / denom[d * HEADS + hh];

  const v4f hv = *(const v4f*)(h + (size_t)s * OUT_DIM + l * 4);
  float* o = out + (size_t)d * OUT_DIM + l * 4;
  atomicAdd(o + 0, alpha * hv.x);
  atomicAdd(o + 1, alpha * hv.y);
  atomicAdd(o + 2, alpha * hv.z);
  atomicAdd(o + 3, alpha * hv.w);
}

// ---------------------------------------------------------------------------
extern "C" void kernel_launch(void* const* d_in, const int* in_sizes, int n_in,
                              void* d_out, int out_size, void* d_ws, size_t ws_size,
                              hipStream_t stream) {
  const float* x       = (const float*)d_in[0];
  const int*   ei      = (const int*)d_in[1];   // [2, E] (src row, dst row)
  const float* W       = (const float*)d_in[2];
  const float* att_src = (const float*)d_in[3];
  const float* att_dst = (const float*)d_in[4];
  const float* bias    = (const float*)d_in[5];
  float*       out     = (float*)d_out;

  const int E = in_sizes[1] / 2;

  // Workspace layout: h [N,256] | asrc [N,4] | adst [N,4] | max [N,4] | den [N,4]
  float* h     = (float*)d_ws;
  float* asrc  = h     + (size_t)N_NODES * OUT_DIM;
  float* adst  = asrc  + (size_t)N_NODES * HEADS;
  float* mmax  = adst  + (size_t)N_NODES * HEADS;
  float* denom = mmax  + (size_t)N_NODES * HEADS;

  // 1) h = x @ W  (WMMA, W staged via LDS)
  {
    int grid = (ROWTILES + WAVES_PER_WG - 1) / WAVES_PER_WG;  // 391
    gat_gemm_wmma<<<grid, 256, 0, stream>>>(x, W, h);
  }

  // 2) attention coefficients + init max/denom
  {
    int n = N_NODES * HEADS;
    gat_attn_coef<<<(n + 255) / 256, 256, 0, stream>>>(h, att_src, att_dst,
                                                       asrc, adst, mmax, denom);
  }

  // 3) out = bias
  {
    int n = N_NODES * OUT_DIM;
    gat_out_init<<<(n + 255) / 256, 256, 0, stream>>>(bias, out);
  }

  // 4) segment max
  {
    int n = (E + N_NODES) * HEADS;
    gat_edge_max<<<(n + 255) / 256, 256, 0, stream>>>(ei, E, asrc, adst, mmax);
  }

  // 5) segment exp-sum
  {
    int n = (E + N_NODES) * HEADS;
    gat_edge_sum<<<(n + 255) / 256, 256, 0, stream>>>(ei, E, asrc, adst, mmax,
                                                      denom);
  }

  // 6) weighted scatter-add
  {
    int n = (E + N_NODES) * 64;
    gat_edge_scatter<<<(n + 255) / 256, 256, 0, stream>>>(ei, E, h, asrc, adst,
                                                          mmax, denom, out);
  }
}